// ConnectionTransformer_2113123910036
// MI455X (gfx1250) — compile-verified
//
#include <hip/hip_runtime.h>
#include <hip/hip_bf16.h>
#include <math.h>

// ---------------- model constants ----------------
#define Bsz 4
#define LS  256
#define LT  256
#define VOCAB 32000
#define Dm  256
#define Sn  128
#define Rr  32
#define NH  8
#define HD  32
#define NLa 6
#define NST 6
#define NEGF (-1.0e9f)

typedef float v2f __attribute__((ext_vector_type(2)));
typedef float v8f __attribute__((ext_vector_type(8)));

// ---- CDNA5 async global->LDS copies (ASYNCcnt-tracked), via inline asm ----
// The LDS base pointer is passed as an (unused) "v" input so the shared array
// escapes into the asm: this stops GlobalOpt folding the never-visibly-stored
// LDS global's loads to undef, and with the "memory" clobber orders the async
// writes against the ds_load fragment reads.  (A "=m" LDS operand is not
// selectable by the backend.)
#define ASYNC_LOAD_B128(base, ldsoff, gaddr) \
  asm volatile("global_load_async_to_lds_b128 %0, %1, off" \
               :: "v"(ldsoff), "v"(gaddr), "v"((const void*)(base)) : "memory")
#define ASYNC_LOAD_B32(base, ldsoff, gaddr) \
  asm volatile("global_load_async_to_lds_b32 %0, %1, off" \
               :: "v"(ldsoff), "v"(gaddr), "v"((const void*)(base)) : "memory")
#define ASYNC_WAIT() asm volatile("s_wait_asynccnt 0x0" ::: "memory")

// =====================================================================
// Generic batched GEMM:  C = act( alpha * A·B + bias )   (WMMA f32 16x16x4)
//   A is M x K  (transA: element(m,k) = A[k*lda+m], else A[m*lda+k])
//   B          (transB: element(k,n) = B[n*ldb+k], else B[k*ldb+n])
//   batch z = zo*inner + zi, pointer offsets via strides.
//   acc=1: C += result (C preloaded into accumulator; alpha must be 1).
//   act=1: exact GELU.
// Block tile 128(M) x 32(N), 256 threads = 8 waves; each wave owns a
// 32x16 region (two stacked 16x16 WMMA accumulators sharing B frags).
// Tiles staged to LDS with GLOBAL_LOAD_ASYNC_TO_LDS.
// =====================================================================
#define AS_PITCH 20
#define BS_BASE  (128 * AS_PITCH)   // floats: A tile 128x20, then B tile 16x36
#define BS_PITCH 36
#define SMEM_N   (BS_BASE + 16 * BS_PITCH)

struct GemmP {
  const float* A; const float* Bm; const float* bias; float* C;
  int M, N, K, lda, ldb, ldc;
  int transA, transB, acc, act;
  float alpha;
  long long sAo, sAi, sBo, sBi, sCo, sCi;
  int innerB;
};

__global__ __launch_bounds__(256) void gemm_wmma_kernel(GemmP p) {
  __shared__ float smem[SMEM_N];   // single LDS block @ offset 0

  const int z  = blockIdx.z;
  const int zo = z / p.innerB, zi = z % p.innerB;
  const float* A  = p.A  + zo * p.sAo + zi * p.sAi;
  const float* Bm = p.Bm + zo * p.sBo + zi * p.sBi;
  float*       C  = p.C  + zo * p.sCo + zi * p.sCi;

  const int m0 = blockIdx.y * 128;
  const int n0 = blockIdx.x * 32;
  const int t    = threadIdx.x;
  const int wave = t >> 5, lane = t & 31;
  const int lm = lane & 15, kh = lane >> 4;
  const int wm = wave >> 1;            // 0..3 : rows wm*32 .. wm*32+31
  const int wn = wave & 1;             // cols wn*16

  const int n = n0 + wn * 16 + lm;
  v8f c0 = {}, c1 = {};
  if (p.acc) {
    const float* Cp0 = C + (long long)(m0 + wm * 32) * p.ldc + n;
    const float* Cp1 = Cp0 + (long long)16 * p.ldc;
#pragma unroll
    for (int v = 0; v < 8; ++v) {
      c0[v] = Cp0[(long long)(v + 8 * kh) * p.ldc];
      c1[v] = Cp1[(long long)(v + 8 * kh) * p.ldc];
    }
  }

  for (int k0 = 0; k0 < p.K; k0 += 16) {
    __syncthreads();
    // ---- stage A tile (128 x 16) into smem[am*20 + ak] ----
    if (!p.transA) {
      const int am = t >> 1, ak = (t & 1) * 8;
      const float* src = A + (long long)(m0 + am) * p.lda + k0 + ak;
      const unsigned lo = (unsigned)((am * AS_PITCH + ak) * 4);
      ASYNC_LOAD_B128(smem, lo, src);
      ASYNC_LOAD_B128(smem, lo + 16u, src + 4);
    } else {
      const int am = t & 127, kb = t >> 7;   // kb in {0,1}
#pragma unroll
      for (int it = 0; it < 8; ++it) {
        const int ak = kb + it * 2;
        const unsigned lo = (unsigned)((am * AS_PITCH + ak) * 4);
        ASYNC_LOAD_B32(smem, lo, A + (long long)(k0 + ak) * p.lda + m0 + am);
      }
    }
    // ---- stage B tile (16 x 32) into smem[BS_BASE + bk*36 + bn] ----
    if (p.transB) {
      const int bk = t & 15, bn = t >> 4;
      ASYNC_LOAD_B32(smem, (unsigned)((BS_BASE + bk * BS_PITCH + bn) * 4),
                     Bm + (long long)(n0 + bn) * p.ldb + k0 + bk);
      ASYNC_LOAD_B32(smem, (unsigned)((BS_BASE + bk * BS_PITCH + bn + 16) * 4),
                     Bm + (long long)(n0 + bn + 16) * p.ldb + k0 + bk);
    } else {
      const int bn = t & 31, bk = t >> 5;
      ASYNC_LOAD_B32(smem, (unsigned)((BS_BASE + bk * BS_PITCH + bn) * 4),
                     Bm + (long long)(k0 + bk) * p.ldb + n0 + bn);
      ASYNC_LOAD_B32(smem, (unsigned)((BS_BASE + (bk + 8) * BS_PITCH + bn) * 4),
                     Bm + (long long)(k0 + bk + 8) * p.ldb + n0 + bn);
    }
    ASYNC_WAIT();
    __syncthreads();

#pragma unroll
    for (int kk = 0; kk < 16; kk += 4) {
      v2f a0, a1, b;
      // ISA f32 layouts: A frag v-th VGPR holds K = kk + 2*laneHalf + v
      b[0]  = smem[BS_BASE + (kk + 2 * kh + 0) * BS_PITCH + wn * 16 + lm];
      b[1]  = smem[BS_BASE + (kk + 2 * kh + 1) * BS_PITCH + wn * 16 + lm];
      a0[0] = smem[(wm * 32 + lm) * AS_PITCH + kk + 2 * kh + 0];
      a0[1] = smem[(wm * 32 + lm) * AS_PITCH + kk + 2 * kh + 1];
      a1[0] = smem[(wm * 32 + 16 + lm) * AS_PITCH + kk + 2 * kh + 0];
      a1[1] = smem[(wm * 32 + 16 + lm) * AS_PITCH + kk + 2 * kh + 1];
      c0 = __builtin_amdgcn_wmma_f32_16x16x4_f32(false, a0, false, b, (short)0, c0, false, false);
      c1 = __builtin_amdgcn_wmma_f32_16x16x4_f32(false, a1, false, b, (short)0, c1, false, false);
    }
  }

  const float bi = p.bias ? p.bias[n] : 0.0f;
  float* Cp0 = C + (long long)(m0 + wm * 32) * p.ldc + n;
  float* Cp1 = Cp0 + (long long)16 * p.ldc;
#pragma unroll
  for (int v = 0; v < 8; ++v) {
    float x = c0[v] * p.alpha + bi;
    float y = c1[v] * p.alpha + bi;
    if (p.act == 1) {
      x = 0.5f * x * (1.0f + erff(x * 0.70710678118654752f));
      y = 0.5f * y * (1.0f + erff(y * 0.70710678118654752f));
    }
    Cp0[(long long)(v + 8 * kh) * p.ldc] = x;
    Cp1[(long long)(v + 8 * kh) * p.ldc] = y;
  }
}

// =====================================================================
// Fused reasoning step (bandwidth-bound; reads each W element once):
//   infl[b,j,:] += sum_i ( Hs[b,i,:] @ W_src[i,j,:,:] ) @ W_tgt[i,j,:,:]
// grid = (j = 0..S-1, chunk = 0..7), 16 i per chunk, 256 threads.
// =====================================================================
#define RSN_N (4 * Dm + 2 * 4 * Rr)

__global__ __launch_bounds__(256) void reason_kernel(
    const float* __restrict__ Hs, const float* __restrict__ Wsrc,
    const float* __restrict__ Wtgt, float* __restrict__ infl) {
  // rsm[0 .. 4*Dm)          : Hs rows for current i (4 x 256)
  // rsm[4*Dm .. 4*Dm+256)   : inter partials [half][b][r]
  __shared__ float rsm[RSN_N];

  const int j = blockIdx.x, ch = blockIdx.y;
  const int t = threadIdx.x;
  const int wave = t >> 5, lane = t & 31;
  const int wb = wave & 3, whalf = wave >> 2;

  float acc0 = 0.f, acc1 = 0.f, acc2 = 0.f, acc3 = 0.f;

  for (int ii = 0; ii < 16; ++ii) {
    const int i = ch * 16 + ii;
    __syncthreads();
    {
      const int b = t >> 6, d = (t & 63) * 4;
      ASYNC_LOAD_B128(rsm, (unsigned)((b * Dm + d) * 4),
                      Hs + ((long long)b * Sn + i) * Dm + d);
    }
    ASYNC_WAIT();
    __syncthreads();
    if (i != j) {   // nondiag factor
      // ---- inter[b][r] = sum_d hs[b][d] * Wsrc[i,j,d,r] ----
      const float* ws = Wsrc + (((long long)i * Sn + j) * Dm + whalf * 128) * Rr + lane;
      __builtin_prefetch(ws + (long long)128 * Rr, 0, 1);
      float s = 0.f;
#pragma unroll 8
      for (int d = 0; d < 128; ++d) s += rsm[wb * Dm + whalf * 128 + d] * ws[(long long)d * Rr];
      rsm[4 * Dm + (whalf * 4 + wb) * Rr + lane] = s;
      __syncthreads();
      // ---- acc[b][d=t] += inter[b][r] * Wtgt[i,j,r,d] ----
      const float* wt = Wtgt + ((long long)i * Sn + j) * Rr * Dm + t;
#pragma unroll 4
      for (int r = 0; r < Rr; ++r) {
        const float w = wt[(long long)r * Dm];
        acc0 += (rsm[4 * Dm + 0 * Rr + r] + rsm[4 * Dm + 4 * Rr + r]) * w;
        acc1 += (rsm[4 * Dm + 1 * Rr + r] + rsm[4 * Dm + 5 * Rr + r]) * w;
        acc2 += (rsm[4 * Dm + 2 * Rr + r] + rsm[4 * Dm + 6 * Rr + r]) * w;
        acc3 += (rsm[4 * Dm + 3 * Rr + r] + rsm[4 * Dm + 7 * Rr + r]) * w;
      }
    }
  }
  atomicAdd(&infl[((long long)0 * Sn + j) * Dm + t], acc0);
  atomicAdd(&infl[((long long)1 * Sn + j) * Dm + t], acc1);
  atomicAdd(&infl[((long long)2 * Sn + j) * Dm + t], acc2);
  atomicAdd(&infl[((long long)3 * Sn + j) * Dm + t], acc3);
}

// ---------------- small utility kernels ----------------
__global__ void fill_kernel(float* p, float v, long long n) {
  long long i = (long long)blockIdx.x * blockDim.x + threadIdx.x;
  if (i < n) p[i] = v;
}
__global__ void add_kernel(float* y, const float* x, long long n) {
  long long i = (long long)blockIdx.x * blockDim.x + threadIdx.x;
  if (i < n) y[i] += x[i];
}
__global__ void embed_kernel(const int* __restrict__ ids, const float* __restrict__ tok,
                             const float* __restrict__ pos, float* __restrict__ out, int L) {
  const int bl = blockIdx.x, t = threadIdx.x;
  const int l = bl % L;
  const int id = ids[bl];
  out[(long long)bl * Dm + t] = tok[(long long)id * Dm + t] + pos[(long long)l * Dm + t];
}
__global__ void copy_slots_kernel(float* __restrict__ Hs, const float* __restrict__ H) {
  const int row = blockIdx.x, t = threadIdx.x;   // row = b*S+s
  Hs[(long long)row * Dm + t] = H[(long long)(row % Sn) * Dm + t];
}

// LayerNorm over D=256, one block per row
__global__ void ln_kernel(const float* __restrict__ x, float* __restrict__ y,
                          const float* __restrict__ g, const float* __restrict__ b) {
  const int row = blockIdx.x, t = threadIdx.x;
  __shared__ float red[256];
  float v = x[(long long)row * Dm + t];
  red[t] = v; __syncthreads();
  for (int o = 128; o > 0; o >>= 1) { if (t < o) red[t] += red[t + o]; __syncthreads(); }
  const float mu = red[0] * (1.0f / Dm); __syncthreads();
  const float dx = v - mu;
  red[t] = dx * dx; __syncthreads();
  for (int o = 128; o > 0; o >>= 1) { if (t < o) red[t] += red[t + o]; __syncthreads(); }
  const float var = red[0] * (1.0f / Dm);
  y[(long long)row * Dm + t] = dx * rsqrtf(var + 1e-5f) * g[t] + b[t];
}

// Hs = LN( Hs + relu(infl) )
__global__ void hs_update_ln_kernel(float* __restrict__ Hs, const float* __restrict__ infl,
                                    const float* __restrict__ g, const float* __restrict__ b) {
  const int row = blockIdx.x, t = threadIdx.x;
  __shared__ float red[256];
  float v = Hs[(long long)row * Dm + t] + fmaxf(infl[(long long)row * Dm + t], 0.0f);
  red[t] = v; __syncthreads();
  for (int o = 128; o > 0; o >>= 1) { if (t < o) red[t] += red[t + o]; __syncthreads(); }
  const float mu = red[0] * (1.0f / Dm); __syncthreads();
  const float dx = v - mu;
  red[t] = dx * dx; __syncthreads();
  for (int o = 128; o > 0; o >>= 1) { if (t < o) red[t] += red[t + o]; __syncthreads(); }
  const float var = red[0] * (1.0f / Dm);
  Hs[(long long)row * Dm + t] = dx * rsqrtf(var + 1e-5f) * g[t] + b[t];
}

// Row softmax; mode 1 = causal + pad mask (self-attn), mode 0 = plain
__global__ void softmax_kernel(float* __restrict__ sc, int cols, int mode,
                               const int* __restrict__ pad, int rowsPerB) {
  const int row = blockIdx.x, t = threadIdx.x;
  float* p = sc + (long long)row * cols;
  __shared__ float red[256];
  const bool on = t < cols;
  float val = -3.0e38f;
  if (on) {
    val = p[t];
    if (mode == 1) {
      const int q = row % LT;
      const int b = row / rowsPerB;
      if (t > q) val += NEGF;
      if (pad[b * LT + t] == 0) val += NEGF;
    }
  }
  red[t] = val; __syncthreads();
  for (int o = 128; o > 0; o >>= 1) { if (t < o) red[t] = fmaxf(red[t], red[t + o]); __syncthreads(); }
  const float mx = red[0]; __syncthreads();
  const float e = on ? __expf(val - mx) : 0.0f;
  red[t] = e; __syncthreads();
  for (int o = 128; o > 0; o >>= 1) { if (t < o) red[t] += red[t + o]; __syncthreads(); }
  const float sum = red[0];
  if (on) p[t] = e / sum;
}

// Slot attention column renorm: A[b,l,s] = A*m / (sum_l(A*m) + 1e-8)
__global__ void slot_colnorm_kernel(float* __restrict__ A, const int* __restrict__ mask) {
  const int b = blockIdx.x / Sn, s = blockIdx.x % Sn;
  const int t = threadIdx.x;   // l (LS == 256 == blockDim)
  __shared__ float red[256];
  const float v = A[((long long)b * LS + t) * Sn + s] * (float)mask[b * LS + t];
  red[t] = v; __syncthreads();
  for (int o = 128; o > 0; o >>= 1) { if (t < o) red[t] += red[t + o]; __syncthreads(); }
  const float sum = red[0] + 1e-8f;
  A[((long long)b * LS + t) * Sn + s] = v / sum;
}

// ---------------- host side ----------------
static void gemm(hipStream_t st, const float* A, const float* B, const float* bias, float* C,
                 int M, int N, int K, int lda, int ldb, int ldc,
                 int tA, int tB, int acc, int act, float alpha,
                 int nz, int inner,
                 long long sAo, long long sAi, long long sBo, long long sBi,
                 long long sCo, long long sCi) {
  GemmP p;
  p.A = A; p.Bm = B; p.bias = bias; p.C = C;
  p.M = M; p.N = N; p.K = K; p.lda = lda; p.ldb = ldb; p.ldc = ldc;
  p.transA = tA; p.transB = tB; p.acc = acc; p.act = act; p.alpha = alpha;
  p.sAo = sAo; p.sAi = sAi; p.sBo = sBo; p.sBi = sBi; p.sCo = sCo; p.sCi = sCi;
  p.innerB = inner < 1 ? 1 : inner;
  dim3 g(N / 32, M / 128, nz), b(256);
  gemm_wmma_kernel<<<g, b, 0, st>>>(p);
}

extern "C" void kernel_launch(void* const* d_in, const int* in_sizes, int n_in,
                              void* d_out, int out_size, void* d_ws, size_t ws_size,
                              hipStream_t stream) {
  (void)in_sizes; (void)n_in; (void)out_size; (void)ws_size;
  const int*   src_ids   = (const int*)d_in[0];
  const int*   tgt_ids   = (const int*)d_in[1];
  const int*   src_mask  = (const int*)d_in[2];
  const int*   tgt_mask  = (const int*)d_in[3];
  const float* src_tok   = (const float*)d_in[4];
  const float* src_pos   = (const float*)d_in[5];
  const float* tgt_tok   = (const float*)d_in[6];
  const float* tgt_pos   = (const float*)d_in[7];
  const float* H_slots   = (const float*)d_in[8];
  const float* W_source  = (const float*)d_in[9];
  const float* W_target  = (const float*)d_in[10];
  const float* Wq_in     = (const float*)d_in[11];
  const float* Wk_slots  = (const float*)d_in[12];
  const float* Wv_in     = (const float*)d_in[13];
  const float* ln_r_g    = (const float*)d_in[14];
  const float* ln_r_b    = (const float*)d_in[15];
  const float* sa_qkv_w  = (const float*)d_in[16];
  const float* sa_qkv_b  = (const float*)d_in[17];
  const float* sa_o_w    = (const float*)d_in[18];
  const float* sa_o_b    = (const float*)d_in[19];
  const float* ca_qkv_w  = (const float*)d_in[20];
  const float* ca_qkv_b  = (const float*)d_in[21];
  const float* ca_o_w    = (const float*)d_in[22];
  const float* ca_o_b    = (const float*)d_in[23];
  const float* ln1_g = (const float*)d_in[24]; const float* ln1_b = (const float*)d_in[25];
  const float* ln2_g = (const float*)d_in[26]; const float* ln2_b = (const float*)d_in[27];
  const float* ln3_g = (const float*)d_in[28]; const float* ln3_b = (const float*)d_in[29];
  const float* ff1_w = (const float*)d_in[30]; const float* ff1_b = (const float*)d_in[31];
  const float* ff2_w = (const float*)d_in[32]; const float* ff2_b = (const float*)d_in[33];
  const float* out_ln_g = (const float*)d_in[34];
  const float* out_ln_b = (const float*)d_in[35];
  const float* out_proj = (const float*)d_in[36];
  float* out = (float*)d_out;

  // workspace bump allocator (floats)
  float* wsp = (float*)d_ws;
  auto alloc = [&](long long n) { float* p = wsp; wsp += n; return p; };
  float* X     = alloc((long long)Bsz * LS * Dm);
  float* Qb0   = alloc((long long)Bsz * LS * Dm);
  float* Ks    = alloc((long long)Sn * Dm);
  float* Vv    = alloc((long long)Bsz * LS * Dm);
  float* Aslot = alloc((long long)Bsz * LS * Sn);
  float* Hs    = alloc((long long)Bsz * Sn * Dm);
  float* infl  = alloc((long long)Bsz * Sn * Dm);
  float* Y     = alloc((long long)Bsz * LT * Dm);
  float* hbuf  = alloc((long long)Bsz * LT * Dm);
  float* qb    = alloc((long long)Bsz * LT * Dm);
  float* kb    = alloc((long long)Bsz * LT * Dm);
  float* vb    = alloc((long long)Bsz * LT * Dm);
  float* sco   = alloc((long long)Bsz * NH * LT * LT);
  float* attno = alloc((long long)Bsz * LT * Dm);
  float* tmp   = alloc((long long)Bsz * LT * Dm);
  float* ffb   = alloc((long long)Bsz * LT * 4 * Dm);

  const int Mx = Bsz * LS;  // 1024 tokens (src == tgt lengths here)
  const float invSqrtD  = 0.0625f;               // 1/sqrt(256)
  const float invSqrtHD = 0.17677669529663688f;  // 1/sqrt(32)

  // ---------- source encoding + slot attention ----------
  embed_kernel<<<Bsz * LS, 256, 0, stream>>>(src_ids, src_tok, src_pos, X, LS);
  gemm(stream, X, Wq_in, nullptr, Qb0, Mx, Dm, Dm, Dm, Dm, Dm, 0, 1, 0, 0, 1.f, 1, 1, 0,0,0,0,0,0);
  gemm(stream, H_slots, Wk_slots, nullptr, Ks, Sn, Dm, Dm, Dm, Dm, Dm, 0, 1, 0, 0, 1.f, 1, 1, 0,0,0,0,0,0);
  gemm(stream, X, Wv_in, nullptr, Vv, Mx, Dm, Dm, Dm, Dm, Dm, 0, 1, 0, 0, 1.f, 1, 1, 0,0,0,0,0,0);
  // A = softmax(Q K^T / sqrt(D)) over slots
  gemm(stream, Qb0, Ks, nullptr, Aslot, Mx, Sn, Dm, Dm, Dm, Sn, 0, 1, 0, 0, invSqrtD, 1, 1, 0,0,0,0,0,0);
  softmax_kernel<<<Bsz * LS, 256, 0, stream>>>(Aslot, Sn, 0, nullptr, 0);
  slot_colnorm_kernel<<<Bsz * Sn, 256, 0, stream>>>(Aslot, src_mask);
  // Hs = H_slots + A^T @ Vv   (batched over b, transA, accumulate)
  copy_slots_kernel<<<Bsz * Sn, 256, 0, stream>>>(Hs, H_slots);
  gemm(stream, Aslot, Vv, nullptr, Hs, Sn, Dm, LS, Sn, Dm, Dm, 1, 0, 1, 0, 1.f,
       Bsz, Bsz, 0, (long long)LS * Sn, 0, (long long)LS * Dm, 0, (long long)Sn * Dm);

  // ---------- reasoning steps ----------
  const long long nHs = (long long)Bsz * Sn * Dm;
  for (int tstep = 0; tstep < NST; ++tstep) {
    fill_kernel<<<(int)((nHs + 255) / 256), 256, 0, stream>>>(infl, 0.f, nHs);
    reason_kernel<<<dim3(Sn, 8), 256, 0, stream>>>(Hs, W_source, W_target, infl);
    hs_update_ln_kernel<<<Bsz * Sn, 256, 0, stream>>>(Hs, infl,
        ln_r_g + (long long)tstep * Dm, ln_r_b + (long long)tstep * Dm);
  }

  // ---------- decoder ----------
  embed_kernel<<<Bsz * LT, 256, 0, stream>>>(tgt_ids, tgt_tok, tgt_pos, Y, LT);
  const long long nY = (long long)Bsz * LT * Dm;

  for (int l = 0; l < NLa; ++l) {
    // ===== self attention =====
    ln_kernel<<<Bsz * LT, 256, 0, stream>>>(Y, hbuf, ln1_g + l * Dm, ln1_b + l * Dm);
    const float* qkvw = sa_qkv_w + (long long)l * 3 * Dm * Dm;
    const float* qkvb = sa_qkv_b + (long long)l * 3 * Dm;
    gemm(stream, hbuf, qkvw,                          qkvb,          qb, Mx, Dm, Dm, Dm, Dm, Dm, 0,1,0,0,1.f, 1,1, 0,0,0,0,0,0);
    gemm(stream, hbuf, qkvw + (long long)Dm * Dm,     qkvb + Dm,     kb, Mx, Dm, Dm, Dm, Dm, Dm, 0,1,0,0,1.f, 1,1, 0,0,0,0,0,0);
    gemm(stream, hbuf, qkvw + (long long)2 * Dm * Dm, qkvb + 2 * Dm, vb, Mx, Dm, Dm, Dm, Dm, Dm, 0,1,0,0,1.f, 1,1, 0,0,0,0,0,0);
    // scores (batched over b*NH): M=LT, N=LT, K=HD
    gemm(stream, qb, kb, nullptr, sco, LT, LT, HD, Dm, Dm, LT, 0, 1, 0, 0, invSqrtHD,
         Bsz * NH, NH, (long long)LT * Dm, HD, (long long)LT * Dm, HD,
         (long long)NH * LT * LT, (long long)LT * LT);
    softmax_kernel<<<Bsz * NH * LT, 256, 0, stream>>>(sco, LT, 1, tgt_mask, NH * LT);
    // O = A @ V : M=LT, N=HD, K=LT
    gemm(stream, sco, vb, nullptr, attno, LT, HD, LT, LT, Dm, Dm, 0, 0, 0, 0, 1.f,
         Bsz * NH, NH, (long long)NH * LT * LT, (long long)LT * LT,
         (long long)LT * Dm, HD, (long long)LT * Dm, HD);
    gemm(stream, attno, sa_o_w + (long long)l * Dm * Dm, sa_o_b + l * Dm, tmp,
         Mx, Dm, Dm, Dm, Dm, Dm, 0, 1, 0, 0, 1.f, 1, 1, 0,0,0,0,0,0);
    add_kernel<<<(int)((nY + 255) / 256), 256, 0, stream>>>(Y, tmp, nY);

    // ===== cross attention (K/V from Hs) =====
    ln_kernel<<<Bsz * LT, 256, 0, stream>>>(Y, hbuf, ln2_g + l * Dm, ln2_b + l * Dm);
    const float* cqkvw = ca_qkv_w + (long long)l * 3 * Dm * Dm;
    const float* cqkvb = ca_qkv_b + (long long)l * 3 * Dm;
    gemm(stream, hbuf, cqkvw, cqkvb, qb, Mx, Dm, Dm, Dm, Dm, Dm, 0,1,0,0,1.f, 1,1, 0,0,0,0,0,0);
    gemm(stream, Hs, cqkvw + (long long)Dm * Dm,     cqkvb + Dm,     kb, Bsz * Sn, Dm, Dm, Dm, Dm, Dm, 0,1,0,0,1.f, 1,1, 0,0,0,0,0,0);
    gemm(stream, Hs, cqkvw + (long long)2 * Dm * Dm, cqkvb + 2 * Dm, vb, Bsz * Sn, Dm, Dm, Dm, Dm, Dm, 0,1,0,0,1.f, 1,1, 0,0,0,0,0,0);
    // scores: M=LT, N=S, K=HD
    gemm(stream, qb, kb, nullptr, sco, LT, Sn, HD, Dm, Dm, Sn, 0, 1, 0, 0, invSqrtHD,
         Bsz * NH, NH, (long long)LT * Dm, HD, (long long)Sn * Dm, HD,
         (long long)NH * LT * Sn, (long long)LT * Sn);
    softmax_kernel<<<Bsz * NH * LT, 256, 0, stream>>>(sco, Sn, 0, nullptr, 0);
    gemm(stream, sco, vb, nullptr, attno, LT, HD, Sn, Sn, Dm, Dm, 0, 0, 0, 0, 1.f,
         Bsz * NH, NH, (long long)NH * LT * Sn, (long long)LT * Sn,
         (long long)Sn * Dm, HD, (long long)LT * Dm, HD);
    gemm(stream, attno, ca_o_w + (long long)l * Dm * Dm, ca_o_b + l * Dm, tmp,
         Mx, Dm, Dm, Dm, Dm, Dm, 0, 1, 0, 0, 1.f, 1, 1, 0,0,0,0,0,0);
    add_kernel<<<(int)((nY + 255) / 256), 256, 0, stream>>>(Y, tmp, nY);

    // ===== FFN (exact GELU) =====
    ln_kernel<<<Bsz * LT, 256, 0, stream>>>(Y, hbuf, ln3_g + l * Dm, ln3_b + l * Dm);
    gemm(stream, hbuf, ff1_w + (long long)l * 4 * Dm * Dm, ff1_b + (long long)l * 4 * Dm, ffb,
         Mx, 4 * Dm, Dm, Dm, Dm, 4 * Dm, 0, 1, 0, 1 /*gelu*/, 1.f, 1, 1, 0,0,0,0,0,0);
    gemm(stream, ffb, ff2_w + (long long)l * Dm * 4 * Dm, ff2_b + (long long)l * Dm, tmp,
         Mx, Dm, 4 * Dm, 4 * Dm, 4 * Dm, Dm, 0, 1, 0, 0, 1.f, 1, 1, 0,0,0,0,0,0);
    add_kernel<<<(int)((nY + 255) / 256), 256, 0, stream>>>(Y, tmp, nY);
  }

  // ---------- output projection ----------
  ln_kernel<<<Bsz * LT, 256, 0, stream>>>(Y, hbuf, out_ln_g, out_ln_b);
  gemm(stream, hbuf, out_proj, nullptr, out, Mx, VOCAB, Dm, Dm, Dm, VOCAB,
       0, 1, 0, 0, 1.f, 1, 1, 0,0,0,0,0,0);
}